// PositionalEncoding_30279519437338
// MI455X (gfx1250) — compile-verified
//
#include <hip/hip_runtime.h>
#include <math.h>

// Problem shape (fixed by the reference): B=64, S=4096, N=25, F=3.
#define NPTS 25u
#define SEQ  4096u
#define PPB  256u                               // points per block (1 point / thread)
#define TILE_PTS    (PPB + NPTS)                // 281 points staged per block
#define TILE_FLOATS (TILE_PTS * 3u)             // 843 floats
#define IN_CHUNKS   ((TILE_FLOATS * 4u + 15u) / 16u)   // 211 x 16B async-load chunks
#define OUT_FLOATS  (PPB * 5u)                  // 1280 floats out per block
#define OUT_CHUNKS  (OUT_FLOATS * 4u / 16u)     // 320 x 16B async-store chunks

__global__ __launch_bounds__(PPB)
void pe_feats_kernel(const float* __restrict__ x, float* __restrict__ out,
                     unsigned totalPts) {
    // Input tile (843 floats, rounded up to whole 16B chunks) + output staging.
    __shared__ __align__(16) float sIn[IN_CHUNKS * 4u + 4u];
    __shared__ __align__(16) float sOut[OUT_FLOATS];

    const unsigned tid        = threadIdx.x;
    const unsigned blockStart = blockIdx.x * PPB;

    // LDS byte offsets: generic-pointer low 32 bits == LDS offset (aperture mapping).
    const unsigned ldsIn  = (unsigned)(size_t)(void*)sIn;
    const unsigned ldsOut = (unsigned)(size_t)(void*)sOut;

    // ---- Stage x tile into LDS with gfx1250 async 128b loads (ASYNCcnt path) ----
    {
        const unsigned long long xBase = (unsigned long long)(size_t)x;
        const unsigned long long last  = xBase + (unsigned long long)totalPts * 12ull - 16ull;
        if (tid < IN_CHUNKS) {
            unsigned long long ga = xBase
                                  + (unsigned long long)blockStart * 12ull
                                  + (unsigned long long)tid * 16ull;
            if (ga > last) ga = last;           // clamp: tail slots are never consumed
            unsigned la = ldsIn + tid * 16u;
            asm volatile("global_load_async_to_lds_b128 %0, %1, off"
                         :: "v"(la), "v"(ga) : "memory");
        }
    }
    asm volatile("s_wait_asynccnt 0x0" ::: "memory");
    __syncthreads();

    // ---- Per-point feature computation out of LDS ----
    {
        const unsigned p  = blockStart + tid;
        const unsigned s  = (p / NPTS) % SEQ;           // sequence index of this point
        const unsigned l3 = tid * 3u;
        const float x0 = sIn[l3 + 0], x1 = sIn[l3 + 1], x2 = sIn[l3 + 2];

        // next step, same joint: +25 points in the tile; last step pairs with itself
        const unsigned n3 = (s == SEQ - 1u) ? l3 : (l3 + NPTS * 3u);
        const float y0 = sIn[n3 + 0], y1 = sIn[n3 + 1], y2 = sIn[n3 + 2];

        const float d0 = y0 - x0, d1 = y1 - x1, d2 = y2 - x2;
        const float dist = sqrtf(d0 * d0 + d1 * d1 + d2 * d2);

        const float dot = x0 * y0 + x1 * y1 + x2 * y2;
        const float la  = sqrtf(x0 * x0 + x1 * x1 + x2 * x2);
        const float lb  = sqrtf(y0 * y0 + y1 * y1 + y2 * y2);
        float ang = acosf(dot / (la * lb));
        ang = (ang == ang) ? ang : 0.0f;                // nan_to_num

        const unsigned o5 = tid * 5u;
        sOut[o5 + 0] = x0;
        sOut[o5 + 1] = x1;
        sOut[o5 + 2] = x2;
        sOut[o5 + 3] = dist;
        sOut[o5 + 4] = ang;
    }
    __syncthreads();

    // ---- Stream output tile with async 128b LDS->global stores ----
    {
        const unsigned long long oBase = (unsigned long long)(size_t)out
                                       + (unsigned long long)blockStart * 20ull;
        {
            unsigned long long ga = oBase + (unsigned long long)tid * 16ull;
            unsigned la = ldsOut + tid * 16u;
            asm volatile("global_store_async_from_lds_b128 %0, %1, off"
                         :: "v"(ga), "v"(la) : "memory");
        }
        if (tid < OUT_CHUNKS - PPB) {                   // remaining 64 chunks
            unsigned long long ga = oBase + (unsigned long long)(tid + PPB) * 16ull;
            unsigned la = ldsOut + (tid + PPB) * 16u;
            asm volatile("global_store_async_from_lds_b128 %0, %1, off"
                         :: "v"(ga), "v"(la) : "memory");
        }
    }
    // S_ENDPGM implies wait-idle, but make the drain explicit.
    asm volatile("s_wait_asynccnt 0x0" ::: "memory");
}

extern "C" void kernel_launch(void* const* d_in, const int* in_sizes, int n_in,
                              void* d_out, int out_size, void* d_ws, size_t ws_size,
                              hipStream_t stream) {
    (void)n_in; (void)out_size; (void)d_ws; (void)ws_size;
    const float* x  = (const float*)d_in[0];
    float* out      = (float*)d_out;
    const unsigned totalPts = (unsigned)(in_sizes[0] / 3);   // 6,553,600
    const unsigned grid     = totalPts / PPB;                // 25,600 (exact)
    pe_feats_kernel<<<dim3(grid), dim3(PPB), 0, stream>>>(x, out, totalPts);
}